// F2Layer_64055142252771
// MI455X (gfx1250) — compile-verified
//
#include <hip/hip_runtime.h>
#include <hip/hip_bf16.h>

// ---------------------------------------------------------------------------
// Gated graph aggregation (GCN-ish, 4 heads x 64 dims)
//   Phase 1: deg/norm via float atomics
//   Phase 2: per-(node,head) gate dots  g[r,0]=<h_r,Wd>, g[r,1]=<h_r,Ws>
//            via V_WMMA_F32_16X16X4_F32 (exact fp32 tensor op), weights
//            staged in LDS to keep the K-loop branch-free.
//   Phase 3: wave-per-edge tanh gate + coalesced scatter (atomic f32 adds)
// ---------------------------------------------------------------------------

#define N_NODES 20000
#define N_EDGES 320000
#define HEAD_NUM 4
#define IN_DIM 64
#define M_ROWS (N_NODES * HEAD_NUM)   // 80000, divisible by 16

typedef __attribute__((ext_vector_type(2))) float v2f;
typedef __attribute__((ext_vector_type(8))) float v8f;

// ---- zero deg + output -----------------------------------------------------
__global__ void zero_kernel(float* __restrict__ deg, float* __restrict__ z,
                            int n_deg, int n_z) {
    int i = blockIdx.x * blockDim.x + threadIdx.x;
    if (i < n_deg) deg[i] = 0.0f;
    for (int j = i; j < n_z; j += gridDim.x * blockDim.x) z[j] = 0.0f;
}

// ---- in-degree -------------------------------------------------------------
__global__ void deg_kernel(const int* __restrict__ dst, float* __restrict__ deg,
                           int E) {
    int e = blockIdx.x * blockDim.x + threadIdx.x;
    if (e < E) atomicAdd(&deg[dst[e]], 1.0f);
}

// ---- norm = clip(deg,1)^-0.5 ----------------------------------------------
__global__ void norm_kernel(const float* __restrict__ deg, float* __restrict__ nrm,
                            int N) {
    int i = blockIdx.x * blockDim.x + threadIdx.x;
    if (i < N) nrm[i] = rsqrtf(fmaxf(deg[i], 1.0f));
}

// ---- gate precompute with FP32 WMMA ---------------------------------------
// A: rows = (node*4+head), 64 f32 each (h is contiguous (N, 4*64))
// B: 64x2, col0 = Wd = Wg[0..63], col1 = Ws = Wg[64..127]
// One wave -> 16 rows. K looped 64/4 = 16 WMMAs, f32 exact accumulate.
__global__ void gate_wmma_kernel(const float* __restrict__ h,
                                 const float* __restrict__ Wg,
                                 float* __restrict__ g) {
    __shared__ float Wlds[2 * IN_DIM];
    for (int i = threadIdx.x; i < 2 * IN_DIM; i += blockDim.x) Wlds[i] = Wg[i];
    __syncthreads();

    int wave = (blockIdx.x * blockDim.x + threadIdx.x) >> 5;
    int lane = threadIdx.x & 31;
    int row0 = wave * 16;
    if (row0 >= M_ROWS) return;              // wave-uniform (exact fit anyway)

    int  m     = lane & 15;                  // A row / B col within tile
    int  khalf = lane >> 4;                  // 0: K pair (k0,k0+1), 1: (k0+2,k0+3)
    bool live  = (m < 2);                    // only B cols 0,1 are real
    int  bcol  = live ? m : 0;               // clamped (safe) LDS column

    const float* Arow = h + (size_t)(row0 + m) * IN_DIM;
    const float* Brow = &Wlds[bcol * IN_DIM];

    v8f c = {};
#pragma unroll
    for (int k0 = 0; k0 < IN_DIM; k0 += 4) {
        int kb = k0 + khalf * 2;             // even -> 8B-aligned ds_load_b64
        v2f a = *(const v2f*)&Arow[kb];
        v2f w = *(const v2f*)&Brow[kb];
        v2f b;
        b.x = live ? w.x : 0.0f;             // branch-free v_cndmask
        b.y = live ? w.y : 0.0f;
        // (neg_a, A, neg_b, B, c_mod, C, reuse_a, reuse_b)
        c = __builtin_amdgcn_wmma_f32_16x16x4_f32(
                false, a, false, b, (short)0, c, false, false);
    }

    // C layout: lane (n=lane&15) = column; VGPR j -> row j (lanes 0-15)
    //           or row j+8 (lanes 16-31). Only columns 0,1 are meaningful.
    if (live) {
        int rbase = row0 + (khalf ? 8 : 0);
#pragma unroll
        for (int j = 0; j < 8; ++j) {
            g[(size_t)(rbase + j) * 2 + m] = c[j];
        }
    }
}

// ---- per-edge gate + scatter ----------------------------------------------
// One wave per edge: 32 lanes sweep the 256 channels in 8 steps (128B/step,
// fully coalesced), float atomic adds into z[dst]. Uniform per-edge scalars
// are forced into SGPRs via readfirstlane.
__global__ void edge_kernel(const float* __restrict__ h,
                            const int* __restrict__ src,
                            const int* __restrict__ dst,
                            const float* __restrict__ g,
                            const float* __restrict__ nrm,
                            const float* __restrict__ bg,
                            float* __restrict__ z, int E) {
    int wave = (blockIdx.x * blockDim.x + threadIdx.x) >> 5;
    int lane = threadIdx.x & 31;
    if (wave >= E) return;

    // per-wave-uniform edge endpoints -> scalar registers
    int s = __builtin_amdgcn_readfirstlane(src[wave]);
    int d = __builtin_amdgcn_readfirstlane(dst[wave]);

    float nn = nrm[s] * nrm[d];
    float b  = bg[0];

    float sh[HEAD_NUM];
#pragma unroll
    for (int hd = 0; hd < HEAD_NUM; ++hd) {
        float gd = g[((size_t)d * HEAD_NUM + hd) * 2 + 0];   // <h[dst],Wd>
        float gs = g[((size_t)s * HEAD_NUM + hd) * 2 + 1];   // <h[src],Ws>
        sh[hd] = tanhf(gd + gs + b) * nn;
    }

    const float* hs = h + (size_t)s * (HEAD_NUM * IN_DIM);
    float*       zd = z + (size_t)d * (HEAD_NUM * IN_DIM);
#pragma unroll
    for (int i = 0; i < 8; ++i) {
        int ch = i * 32 + lane;                    // channel 0..255
        atomicAdd(&zd[ch], hs[ch] * sh[i >> 1]);   // head = ch>>6 == i>>1
    }
}

// ---------------------------------------------------------------------------
extern "C" void kernel_launch(void* const* d_in, const int* in_sizes, int n_in,
                              void* d_out, int out_size, void* d_ws, size_t ws_size,
                              hipStream_t stream) {
    const float* h   = (const float*)d_in[0];   // (N, 256) f32
    const float* Wg  = (const float*)d_in[1];   // (1, 128) f32
    const float* bg  = (const float*)d_in[2];   // (1,) f32
    const int*   src = (const int*)d_in[3];     // (E,)
    const int*   dst = (const int*)d_in[4];     // (E,)
    float*       z   = (float*)d_out;           // (N, 256) f32

    float* ws  = (float*)d_ws;
    float* deg = ws;                            // N floats
    float* nrm = ws + N_NODES;                  // N floats
    float* g   = ws + 2 * N_NODES;              // M_ROWS*2 floats

    const int n_z = N_NODES * HEAD_NUM * IN_DIM;

    // 1) zero deg + z
    {
        int threads = 256;
        int blocks  = (n_z + threads - 1) / threads;   // covers both arrays
        zero_kernel<<<blocks, threads, 0, stream>>>(deg, z, N_NODES, n_z);
    }
    // 2) degree
    deg_kernel<<<(N_EDGES + 255) / 256, 256, 0, stream>>>(dst, deg, N_EDGES);
    // 3) norm
    norm_kernel<<<(N_NODES + 255) / 256, 256, 0, stream>>>(deg, nrm, N_NODES);
    // 4) WMMA gate precompute: M_ROWS/16 = 5000 waves = 160000 threads
    {
        int waves   = M_ROWS / 16;              // exact
        int threads = 256;                      // 8 waves/block
        int blocks  = (waves * 32 + threads - 1) / threads;
        gate_wmma_kernel<<<blocks, threads, 0, stream>>>(h, Wg, g);
    }
    // 5) edge scatter: one wave per edge
    {
        int threads = 256;
        long long tot = (long long)N_EDGES * 32;
        int blocks  = (int)((tot + threads - 1) / threads);
        edge_kernel<<<blocks, threads, 0, stream>>>(h, src, dst, g, nrm, bg, z,
                                                    N_EDGES);
    }
}